// Encoder_27066883899544
// MI455X (gfx1250) — compile-verified
//
#include <hip/hip_runtime.h>
#include <hip/hip_bf16.h>

// ---------------------------------------------------------------------------
// VGAE GraphConv encoder for MI455X (gfx1250, wave32).
// Scatter-mean aggregation with global_atomic_add_f32 (L2-resident), dense
// parts via V_WMMA_F32_16X16X4_F32 keeping exact fp32 semantics.
// ---------------------------------------------------------------------------

#define NNODES 50000
#define NEDGES 800000

typedef __attribute__((ext_vector_type(2))) float v2f;
typedef __attribute__((ext_vector_type(8))) float v8f;

__global__ void zero_f32(float* __restrict__ p, int n) {
  int i = blockIdx.x * blockDim.x + threadIdx.x;
  if (i < n) p[i] = 0.0f;
}

__global__ void deg_count(const int* __restrict__ dst, float* __restrict__ deg, int nE) {
  int e = blockIdx.x * blockDim.x + threadIdx.x;
  if (e < nE) atomicAdd(&deg[dst[e]], 1.0f);
}

__global__ void deg_invert(const float* __restrict__ deg, float* __restrict__ dinv, int n) {
  int i = blockIdx.x * blockDim.x + threadIdx.x;
  if (i < n) {
    float d = deg[i];
    dinv[i] = (d > 0.0f) ? (1.0f / d) : 0.0f;
  }
}

// One thread handles 4 channels of one edge: float4 gather + 4 float atomics.
__global__ void scatter_add(const int* __restrict__ src, const int* __restrict__ dst,
                            const float* __restrict__ x, float* __restrict__ agg,
                            int nE, int c4 /* = C/4 */) {
  int i = blockIdx.x * blockDim.x + threadIdx.x;
  if (i >= nE * c4) return;
  int e = i / c4;
  int g = i - e * c4;
  int C = c4 * 4;
  const float4 v = reinterpret_cast<const float4*>(x + (long long)src[e] * C)[g];
  float* a = agg + (long long)dst[e] * C + g * 4;
  atomicAdd(a + 0, v.x);
  atomicAdd(a + 1, v.y);
  atomicAdd(a + 2, v.z);
  atomicAdd(a + 3, v.w);
}

// Fused GraphConv GEMM: out = act( (dinv*agg) @ Wrel + bias + x @ Wroot )
// One wave32 computes a 16 x (NSUB*16) fp32 tile with V_WMMA_F32_16X16X4_F32.
// A-frag (16x4 fp32): lane L<16 -> row m0+L, K pair {k0, k0+1}; lane L>=16 -> {k0+2, k0+3}.
// B-frag (4x16): lane indexes column, VGPRs/halves index K symmetrically.
// C/D: acc[j] = row m0 + j + 8*half, col n0 + 16*s + r.
template <int NSUB, bool RELU>
__global__ void __launch_bounds__(256)
graphconv_gemm(const float* __restrict__ agg, const float* __restrict__ dinv,
               const float* __restrict__ Wrel,
               const float* __restrict__ xin, const float* __restrict__ Wroot,
               const float* __restrict__ bias,
               float* __restrict__ out, int K, int N, int nTilesN, int totalWaves) {
  const int wave = blockIdx.x * (blockDim.x >> 5) + (threadIdx.x >> 5);
  if (wave >= totalWaves) return;  // wave-uniform: EXEC stays all-ones for WMMA

  const int lane = threadIdx.x & 31;
  const int half = lane >> 4;
  const int r    = lane & 15;

  const int mT = wave / nTilesN;
  const int nT = wave - mT * nTilesN;
  const int m0 = mT * 16;
  const int n0 = nT * (NSUB * 16);
  const int row = m0 + r;

  const float dv = dinv[row];

  v8f acc[NSUB];
#pragma unroll
  for (int s = 0; s < NSUB; ++s) {
    const float bv = bias[n0 + 16 * s + r];
#pragma unroll
    for (int j = 0; j < 8; ++j) acc[s][j] = bv;
  }

  // Pass 1: mean-aggregated neighbors times W_rel (deg_inv folded into A-frag).
  {
    const float* arow = agg + (long long)row * K;
    for (int k0 = 0; k0 < K; k0 += 4) {
      const int ka = k0 + 2 * half;
      v2f a;
      a.x = arow[ka] * dv;
      a.y = arow[ka + 1] * dv;
#pragma unroll
      for (int s = 0; s < NSUB; ++s) {
        const int col = n0 + 16 * s + r;
        v2f b;
        b.x = Wrel[(long long)ka * N + col];
        b.y = Wrel[(long long)(ka + 1) * N + col];
        acc[s] = __builtin_amdgcn_wmma_f32_16x16x4_f32(
            false, a, false, b, (short)0, acc[s], false, false);
      }
    }
  }

  // Pass 2: root features times W_root, same accumulators.
  {
    const float* xrow = xin + (long long)row * K;
    for (int k0 = 0; k0 < K; k0 += 4) {
      const int ka = k0 + 2 * half;
      v2f a;
      a.x = xrow[ka];
      a.y = xrow[ka + 1];
#pragma unroll
      for (int s = 0; s < NSUB; ++s) {
        const int col = n0 + 16 * s + r;
        v2f b;
        b.x = Wroot[(long long)ka * N + col];
        b.y = Wroot[(long long)(ka + 1) * N + col];
        acc[s] = __builtin_amdgcn_wmma_f32_16x16x4_f32(
            false, a, false, b, (short)0, acc[s], false, false);
      }
    }
  }

#pragma unroll
  for (int s = 0; s < NSUB; ++s) {
#pragma unroll
    for (int j = 0; j < 8; ++j) {
      float v = acc[s][j];
      if (RELU) v = fmaxf(v, 0.0f);
      out[(long long)(m0 + j + 8 * half) * N + n0 + 16 * s + r] = v;
    }
  }
}

static inline int cdiv(long long a, long long b) { return (int)((a + b - 1) / b); }

extern "C" void kernel_launch(void* const* d_in, const int* in_sizes, int n_in,
                              void* d_out, int out_size, void* d_ws, size_t ws_size,
                              hipStream_t stream) {
  // setup_inputs() dict order:
  const float* x       = (const float*)d_in[0];
  const float* W1_rel  = (const float*)d_in[1];
  const float* b1      = (const float*)d_in[2];
  const float* W1_root = (const float*)d_in[3];
  const float* W2_rel  = (const float*)d_in[4];
  const float* b2      = (const float*)d_in[5];
  const float* W2_root = (const float*)d_in[6];
  const float* Wmu_rel = (const float*)d_in[7];
  const float* bmu     = (const float*)d_in[8];
  const float* Wmu_root= (const float*)d_in[9];
  const float* Wls_rel = (const float*)d_in[10];
  const float* bls     = (const float*)d_in[11];
  const float* Wls_root= (const float*)d_in[12];
  const int*   eidx    = (const int*)d_in[13];  // [2, E] flat
  const int* src = eidx;
  const int* dst = eidx + NEDGES;

  float* mu_out = (float*)d_out;                 // 50000*16
  float* ls_out = (float*)d_out + NNODES * 16;   // 50000*16

  // Workspace carve-up (16B-aligned blocks).
  float* ws   = (float*)d_ws;
  float* deg  = ws;                      // 50048
  float* dinv = deg  + 50048;            // 50048
  float* agg  = dinv + 50048;            // up to 50000*256
  float* h1   = agg  + NNODES * 256;     // 50000*256
  float* h2   = h1   + NNODES * 256;     // 50000*128
  (void)ws_size; (void)n_in; (void)in_sizes; (void)out_size;

  const int T = 256;

  // Degree + inverse.
  zero_f32<<<cdiv(NNODES, T), T, 0, stream>>>(deg, NNODES);
  deg_count<<<cdiv(NEDGES, T), T, 0, stream>>>(dst, deg, NEDGES);
  deg_invert<<<cdiv(NNODES, T), T, 0, stream>>>(deg, dinv, NNODES);

  // ---- Layer 1: K=64 -> N=256, ReLU ----
  zero_f32<<<cdiv((long long)NNODES * 64, T), T, 0, stream>>>(agg, NNODES * 64);
  scatter_add<<<cdiv((long long)NEDGES * 16, T), T, 0, stream>>>(src, dst, x, agg, NEDGES, 16);
  {
    const int nTilesN = 256 / 64;              // NSUB=4 -> 64-wide per wave
    const int waves = (NNODES / 16) * nTilesN; // 12500
    graphconv_gemm<4, true><<<cdiv(waves, 8), T, 0, stream>>>(
        agg, dinv, W1_rel, x, W1_root, b1, h1, 64, 256, nTilesN, waves);
  }

  // ---- Layer 2: K=256 -> N=128, ReLU ----
  zero_f32<<<cdiv((long long)NNODES * 256, T), T, 0, stream>>>(agg, NNODES * 256);
  scatter_add<<<cdiv((long long)NEDGES * 64, T), T, 0, stream>>>(src, dst, h1, agg, NEDGES, 64);
  {
    const int nTilesN = 128 / 64;
    const int waves = (NNODES / 16) * nTilesN; // 6250
    graphconv_gemm<4, true><<<cdiv(waves, 8), T, 0, stream>>>(
        agg, dinv, W2_rel, h1, W2_root, b2, h2, 256, 128, nTilesN, waves);
  }

  // ---- Heads: shared aggregation of h2 (K=128 -> N=16), no ReLU ----
  zero_f32<<<cdiv((long long)NNODES * 128, T), T, 0, stream>>>(agg, NNODES * 128);
  scatter_add<<<cdiv((long long)NEDGES * 32, T), T, 0, stream>>>(src, dst, h2, agg, NEDGES, 32);
  {
    const int waves = NNODES / 16; // 3125, NSUB=1 -> 16-wide
    graphconv_gemm<1, false><<<cdiv(waves, 8), T, 0, stream>>>(
        agg, dinv, Wmu_rel, h2, Wmu_root, bmu, mu_out, 128, 16, 1, waves);
    graphconv_gemm<1, false><<<cdiv(waves, 8), T, 0, stream>>>(
        agg, dinv, Wls_rel, h2, Wls_root, bls, ls_out, 128, 16, 1, waves);
  }
}